// Memory_18545668784771
// MI455X (gfx1250) — compile-verified
//
#include <hip/hip_runtime.h>

// Problem sizes (fixed by the reference)
constexpr int kB = 64;
constexpr int kS = 1024;

typedef __attribute__((ext_vector_type(2))) float v2f;
typedef __attribute__((ext_vector_type(4))) float v4f;
typedef __attribute__((ext_vector_type(8))) float v8f;

__device__ __forceinline__ float sigmoidf(float t) {
    return 1.0f / (1.0f + __expf(-t));
}

// One wave (32 lanes) handles one 16-row block of one batch's W1.
//   Phase 1: y-dots via V_WMMA_F32_16X16X4_F32 over K=1024
//   Phase 2: W1_new = y*x^T - decay*W1, coalesced b128 re-read (L2 hit) + store
__global__ __launch_bounds__(256) void fastweight_kernel(
    const float* __restrict__ x,      // [B,S]
    const float* __restrict__ W1,     // [B,S,S]
    const float* __restrict__ delta,  // [S]
    const float* __restrict__ eta,    // [S]
    const float* __restrict__ theta,  // [S]
    float* __restrict__ outY,         // [B,S]
    float* __restrict__ outW)         // [B,S,S]
{
    const int lane = threadIdx.x & 31;
    const int wave = threadIdx.x >> 5;
    const int rb   = blockIdx.x * 8 + wave;   // row-block id: 0 .. B*(S/16)-1
    const int b    = rb >> 6;                 // S/16 == 64 row-blocks per batch
    const int i0   = (rb & 63) << 4;          // first row of this 16-row block

    const float* xb = x  + (size_t)b * kS;
    const float* Wb = W1 + ((size_t)b * kS + i0) * (size_t)kS;

    // ---------------- Phase 1: 16 row-dots via WMMA ----------------
    // A layout (32-bit 16x4): lanes 0-15 hold (M=lane, K=0,1); lanes 16-31 hold (M=lane-16, K=2,3)
    // B layout (4x16): lane holds B[K=koff..koff+1][N=lane&15]; we want B[k][n]=x[k] for all n,
    // which is the identical float2 load of x in every lane.
    const int   M    = lane & 15;
    const int   koff = (lane < 16) ? 0 : 2;
    const float* arow = Wb + (size_t)M * kS + koff;

    v8f c = {};
    #pragma unroll 8
    for (int k = 0; k < kS; k += 4) {
        v2f a  = *(const v2f*)(arow + k);
        v2f bv = *(const v2f*)(xb + k + koff);
        c = __builtin_amdgcn_wmma_f32_16x16x4_f32(
                /*neg_a=*/false, a, /*neg_b=*/false, bv,
                /*c_mod=*/(short)0, c, /*reuse_a=*/false, /*reuse_b=*/false);
    }

    // D layout: row m (0..7)  -> c[m]   in lanes 0-15 (all columns equal)
    //           row m (8..15) -> c[m-8] in lanes 16-31
    float yv[16], dec[16];
    float myY = 0.0f;
    #pragma unroll
    for (int m = 0; m < 16; ++m) {
        float dot = __shfl(c[m & 7], (m < 8) ? 0 : 16, 32);
        int   i   = i0 + m;
        float st  = sigmoidf(theta[i]);
        float se  = sigmoidf(eta[i]);
        float v   = st * xb[i] + dot;
        v         = fminf(fmaxf(v, 0.0f), 6.0f);   // relu6
        yv[m]     = se * v;
        dec[m]    = 1.0f - sigmoidf(delta[i]);
        if (lane == m) myY = yv[m];
    }
    if (lane < 16) outY[(size_t)b * kS + i0 + lane] = myY;

    // ---------------- Phase 2: rank-1 update, coalesced b128 ----------------
    // Preload x[b,:] into registers: 8 chunks of float4 per lane (covers 1024).
    v4f xr[8];
    #pragma unroll
    for (int t = 0; t < 8; ++t)
        xr[t] = *(const v4f*)(xb + t * 128 + lane * 4);

    #pragma unroll
    for (int m = 0; m < 16; ++m) {
        const v4f* wr = (const v4f*)(Wb + (size_t)m * kS) + lane;
        v4f*       wo = (v4f*)(outW + ((size_t)b * kS + i0 + m) * (size_t)kS) + lane;
        const float ym = yv[m];
        const float dm = dec[m];
        #pragma unroll
        for (int t = 0; t < 8; ++t) {
            v4f w = wr[t * 32];          // L2-resident re-read of this block
            wo[t * 32] = ym * xr[t] - dm * w;
        }
    }
}

extern "C" void kernel_launch(void* const* d_in, const int* in_sizes, int n_in,
                              void* d_out, int out_size, void* d_ws, size_t ws_size,
                              hipStream_t stream) {
    const float* x     = (const float*)d_in[0];
    const float* W1    = (const float*)d_in[1];
    const float* delta = (const float*)d_in[2];
    const float* eta   = (const float*)d_in[3];
    const float* theta = (const float*)d_in[4];

    float* outY = (float*)d_out;                 // [B,S] first in tuple order
    float* outW = outY + (size_t)kB * kS;        // [B,S,S] second

    const int rowBlocks = kB * (kS / 16);        // 4096 waves of work
    dim3 grid(rowBlocks / 8);                    // 8 waves (256 threads) per block
    fastweight_kernel<<<grid, 256, 0, stream>>>(x, W1, delta, eta, theta, outY, outW);
}